// Group_60541859004641
// MI455X (gfx1250) — compile-verified
//
#include <hip/hip_runtime.h>

#define B_  16
#define N_  16384
#define G_  512
#define K_  32
#define GT  16      // centers per WMMA tile (M)
#define PT  16      // points per WMMA tile (N)
#define UNR 4       // point tiles per loop iteration (4 WMMAs back-to-back)
#define QCAP (PT * UNR)
#define FPT 256     // FPS threads per block
#define PPT (N_ / FPT)

typedef __attribute__((ext_vector_type(2))) float v2f;
typedef __attribute__((ext_vector_type(8))) float v8f;

// ---------------------------------------------------------------------------
// Kernel 1: farthest point sampling. One block per batch. Running min-dist
// kept in registers (PPT floats/thread); argmax via LDS tree reduction with
// lowest-index tie-break (matches jnp.argmax).
// ---------------------------------------------------------------------------
__global__ void fps_kernel(const float* __restrict__ xyz,
                           float* __restrict__ centers)   // (B,G,3) region of d_out
{
    const int b   = blockIdx.x;
    const int tid = threadIdx.x;
    const float* p = xyz + (size_t)b * N_ * 3;

    __shared__ float rv[FPT];
    __shared__ int   ri[FPT];
    __shared__ float s_c[3];
    __shared__ int   s_far;

    float dloc[PPT];
    #pragma unroll
    for (int k = 0; k < PPT; ++k) dloc[k] = 1e10f;

    if (tid == 0) s_far = 0;
    __syncthreads();

    for (int g = 0; g < G_; ++g) {
        if (tid == 0) {
            const int far = s_far;
            const float cx = p[far * 3 + 0];
            const float cy = p[far * 3 + 1];
            const float cz = p[far * 3 + 2];
            s_c[0] = cx; s_c[1] = cy; s_c[2] = cz;
            float* co = centers + ((size_t)b * G_ + g) * 3;
            co[0] = cx; co[1] = cy; co[2] = cz;
        }
        __syncthreads();
        const float cx = s_c[0], cy = s_c[1], cz = s_c[2];

        float bestv = -1.0f; int besti = 0;
        #pragma unroll 4
        for (int k = 0; k < PPT; ++k) {
            const int i = k * FPT + tid;                 // coalesced
            const float dx = p[i * 3 + 0] - cx;
            const float dy = p[i * 3 + 1] - cy;
            const float dz = p[i * 3 + 2] - cz;
            const float dist = dx * dx + dy * dy + dz * dz;
            const float dm = fminf(dloc[k], dist);
            dloc[k] = dm;
            if (dm > bestv) { bestv = dm; besti = i; }   // first max kept
        }
        rv[tid] = bestv; ri[tid] = besti;
        __syncthreads();
        for (int s = FPT / 2; s > 0; s >>= 1) {
            if (tid < s) {
                const float ov = rv[tid + s]; const int oi = ri[tid + s];
                if (ov > rv[tid] || (ov == rv[tid] && oi < ri[tid])) {
                    rv[tid] = ov; ri[tid] = oi;
                }
            }
            __syncthreads();
        }
        if (tid == 0) s_far = ri[0];
        __syncthreads();
    }
}

// ---------------------------------------------------------------------------
// Kernel 2: KNN via 4x V_WMMA_F32_16X16X4_F32 per iteration + parallel
// threshold-filter into per-center LDS queues (ds atomics) + tiny serial
// insertion drain, then gather + center-subtract.
// One wave (32 lanes) per (batch, 16-center tile).
// Score s = |x|^2 - 2<c,x> = d2 - c2  (c2 is a per-row constant -> same order).
// Filter is conservative (s <= worst); the cold drain applies the exact
// lexicographic (score, index) order, so the final top-K is unique and
// independent of queue arrival order -> deterministic, matches
// jax.lax.top_k tie-breaking (first index wins).
// ---------------------------------------------------------------------------
__global__ void knn_group_kernel(const float* __restrict__ xyz,
                                 const float* __restrict__ centers,
                                 float* __restrict__ nbh,      // (B,G,K,3)
                                 float* __restrict__ idx_out)  // (B,G,K) as float
{
    const int blk   = blockIdx.x;                 // B_*(G_/GT) blocks
    const int b     = blk / (G_ / GT);
    const int gbase = (blk % (G_ / GT)) * GT;
    const int lane  = threadIdx.x;                // 0..31, one full wave
    const bool hi   = (lane & 16) != 0;           // lanes 16-31
    const int m     = lane & 15;

    __shared__ float ld[GT][K_];
    __shared__ int   li[GT][K_];
    __shared__ int   qcnt[GT];
    __shared__ float qd[GT][QCAP];
    __shared__ int   qi[GT][QCAP];

    const float* p = xyz + (size_t)b * N_ * 3;

    for (int k = lane; k < GT * K_; k += 32) {
        (&ld[0][0])[k] = 3.0e38f;
        (&li[0][0])[k] = 0;
    }

    // A matrix 16x4 (ISA layout: lanes 0-15 hold K=0,1; lanes 16-31 K=2,3).
    // Row m of A = (-2cx, -2cy, -2cz, 1); B column n = (x, y, z, |x|^2).
    const float* c0 = centers + ((size_t)b * G_ + gbase + m) * 3;
    const float ccx = c0[0], ccy = c0[1], ccz = c0[2];
    v2f a;
    a.x = hi ? (-2.0f * ccz) : (-2.0f * ccx);
    a.y = hi ? 1.0f          : (-2.0f * ccy);
    __syncthreads();

    for (int t = 0; t < N_; t += PT * UNR) {
        v8f acc[UNR];
        #pragma unroll
        for (int u = 0; u < UNR; ++u) {
            const float* q = p + (size_t)(t + u * PT + m) * 3;
            const float qx = q[0], qy = q[1], qz = q[2];
            v2f bm;
            bm.x = hi ? qz : qx;                                   // v_cndmask
            bm.y = hi ? (qx * qx + qy * qy + qz * qz) : qy;        // v_cndmask
            v8f cz8 = {};
            acc[u] = __builtin_amdgcn_wmma_f32_16x16x4_f32(
                         false, a, false, bm, (short)0, cz8, false, false);
        }
        if (t + PT * UNR < N_)
            __builtin_prefetch(p + (size_t)(t + PT * UNR + m) * 3, 0, 1);

        if (lane < GT) qcnt[lane] = 0;
        __syncthreads();

        // Parallel threshold filter: lane holds D[M=j+8*hi][N=m] in acc[u][j].
        // Threshold ld[M][31] is drain-invariant -> hoisted per row (8 LDS
        // loads/lane/iter). Accept-on-equal keeps the filter exec-simple.
        #pragma unroll
        for (int j = 0; j < 8; ++j) {
            const int M = j + (hi ? 8 : 0);
            const float wd = ld[M][K_ - 1];
            #pragma unroll
            for (int u = 0; u < UNR; ++u) {
                const float s = acc[u][j];
                if (s <= wd) {
                    const int pi = t + u * PT + m;
                    const int pos = atomicAdd(&qcnt[M], 1);   // ds_add_rtn_u32
                    qd[M][pos] = s;
                    qi[M][pos] = pi;
                }
            }
        }
        __syncthreads();

        // Drain: owner lane m serially inserts its (rare) survivors, sorted
        // ascending by (score, index). Keep loops rolled: this path is cold.
        if (!hi) {
            const int cnt = qcnt[m];
            #pragma clang loop unroll(disable)
            for (int c = 0; c < cnt; ++c) {
                const float s = qd[m][c];
                const int   pi = qi[m][c];
                const float wd = ld[m][K_ - 1];
                if (s < wd || (s == wd && pi < li[m][K_ - 1])) {
                    int j = K_ - 1;
                    #pragma clang loop unroll(disable)
                    for (; j > 0; --j) {
                        const float pv = ld[m][j - 1];
                        const int   px = li[m][j - 1];
                        if (!(pv > s || (pv == s && px > pi))) break;
                        ld[m][j] = pv;
                        li[m][j] = px;
                    }
                    ld[m][j] = s; li[m][j] = pi;
                }
            }
        }
        __syncthreads();
    }

    // Gather neighbors, subtract center, emit indices.
    if (!hi) {
        const int g = gbase + m;
        float* nb = nbh + ((size_t)b * G_ + g) * K_ * 3;
        float* io = idx_out + ((size_t)b * G_ + g) * K_;
        #pragma clang loop unroll(disable)
        for (int k = 0; k < K_; ++k) {
            const int pi = li[m][k];
            io[k] = (float)pi;
            const float* q = p + (size_t)pi * 3;
            nb[k * 3 + 0] = q[0] - ccx;
            nb[k * 3 + 1] = q[1] - ccy;
            nb[k * 3 + 2] = q[2] - ccz;
        }
    }
}

// ---------------------------------------------------------------------------
// d_out layout (flat, return order): neighborhood (B,G,K,3) | centers (B,G,3)
// | idx (B,G,K).
// ---------------------------------------------------------------------------
extern "C" void kernel_launch(void* const* d_in, const int* in_sizes, int n_in,
                              void* d_out, int out_size, void* d_ws, size_t ws_size,
                              hipStream_t stream) {
    (void)in_sizes; (void)n_in; (void)out_size; (void)d_ws; (void)ws_size;

    const float* xyz = (const float*)d_in[0];
    float* out = (float*)d_out;

    const size_t NBH_ELEMS = (size_t)B_ * G_ * K_ * 3;   // 786432
    const size_t CTR_ELEMS = (size_t)B_ * G_ * 3;        // 24576

    float* nbh  = out;
    float* ctr  = out + NBH_ELEMS;
    float* idxf = out + NBH_ELEMS + CTR_ELEMS;

    fps_kernel<<<B_, FPT, 0, stream>>>(xyz, ctr);
    knn_group_kernel<<<B_ * (G_ / GT), 32, 0, stream>>>(xyz, ctr, nbh, idxf);
}